// MidLoss_43181601194360
// MI455X (gfx1250) — compile-verified
//
#include <hip/hip_runtime.h>
#include <cstdint>

// ---------------------------------------------------------------------------
// Segment-mean MSE, single HBM pass.
//   per segment s:  val_s = sum_i ||x_i||^2  -  ||sum_i x_i||^2 / len_s
//   out = (1/(N*D)) * sum_s val_s
// One workgroup (8 wave32) per segment. Rows staged global->LDS with CDNA5
// async loads (double buffered); column sums done on the matrix engine via
// V_WMMA_F32_16X16X4_F32 with an all-ones A matrix.
// ---------------------------------------------------------------------------

typedef float v2f __attribute__((ext_vector_type(2)));
typedef float v8f __attribute__((ext_vector_type(8)));
typedef int   v4i __attribute__((ext_vector_type(4)));

typedef __attribute__((address_space(1))) v4i* glb_v4i_ptr;
typedef __attribute__((address_space(3))) v4i* lds_v4i_ptr;

#define DIMS 128
#define TILE_ROWS 32          // 32 rows * 512 B = 16 KB per buffer, x2 buffers

// ---- CDNA5 async global->LDS helpers --------------------------------------
__device__ __forceinline__ void async_wait0() {
#if __has_builtin(__builtin_amdgcn_s_wait_asynccnt)
  __builtin_amdgcn_s_wait_asynccnt(0);
#else
  asm volatile("s_wait_asynccnt 0" ::: "memory");
#endif
}

__device__ __forceinline__ void async_copy16(const float* g, float* l) {
#if __has_builtin(__builtin_amdgcn_global_load_async_to_lds_b128)
  __builtin_amdgcn_global_load_async_to_lds_b128(
      (glb_v4i_ptr)(void*)g,
      (lds_v4i_ptr)(void*)l,
      /*offset=*/0, /*cpol=*/0);
#else
  // Fallback: synchronous 16B copy through VGPRs (still one pass over HBM).
  *(float4*)l = *(const float4*)g;
#endif
}

__device__ __forceinline__ v8f wmma_ones_acc(v2f ones, float b0, float b1, v8f c) {
  v2f b; b[0] = b0; b[1] = b1;
  return __builtin_amdgcn_wmma_f32_16x16x4_f32(
      /*neg_a=*/false, ones, /*neg_b=*/false, b,
      /*c_mod=*/(short)0, c, /*reuse_a=*/false, /*reuse_b=*/false);
}

// ---- kernel 1: exclusive scan of lengths -> row offsets --------------------
__global__ __launch_bounds__(256) void seg_scan_kernel(
    const long long* __restrict__ lens, int S, int* __restrict__ offs) {
  __shared__ int sh[256];
  const int t = threadIdx.x;
  const int chunk = (S + 255) / 256;
  const int base = t * chunk;
  long long tot = 0;
  for (int j = 0; j < chunk; ++j) {
    int i = base + j;
    if (i < S) tot += lens[i];
  }
  sh[t] = (int)tot;
  __syncthreads();
  if (t == 0) {            // 256 serial adds: cheap, deterministic
    int run = 0;
    for (int i = 0; i < 256; ++i) { int v = sh[i]; sh[i] = run; run += v; }
  }
  __syncthreads();
  int run = sh[t];
  for (int j = 0; j < chunk; ++j) {
    int i = base + j;
    if (i < S) { offs[i] = run; run += (int)lens[i]; }
  }
}

// ---- kernel 2: one workgroup per segment ----------------------------------
__global__ __launch_bounds__(256) void seg_mse_kernel(
    const float* __restrict__ in, const long long* __restrict__ lens,
    const int* __restrict__ offs, float* __restrict__ segval) {
  __shared__ float buf[2][TILE_ROWS * DIMS];   // 2 x 16 KB streaming buffers
  __shared__ float red_col[DIMS];              // 128 column sums
  __shared__ float red_sq[256];                // per-thread sumsq partials

  const int seg  = blockIdx.x;
  const int tid  = threadIdx.x;
  const int wave = tid >> 5;                   // 0..7, owns 16-column slab
  const int lane = tid & 31;
  const int khalf = lane >> 4;                 // lane halves hold different K rows
  const int n16   = lane & 15;                 // B/C/D column within slab
  const int col   = wave * 16 + n16;           // global column 0..127

  const int len = (int)lens[seg];
  const long long off = offs[seg];
  const int ntiles = (len + TILE_ROWS - 1) / TILE_ROWS;
  const int rows_per_wave = TILE_ROWS / 8;     // 4 rows staged per wave

  // stage tile 0 (each lane moves 16B; one wave op per 512B row)
  {
    const float* gbase = in + off * (long long)DIMS;
    for (int rr = 0; rr < rows_per_wave; ++rr) {
      int r = wave * rows_per_wave + rr;
      if (r < len)
        async_copy16(gbase + (long long)r * DIMS + lane * 4,
                     &buf[0][r * DIMS + lane * 4]);
    }
  }
  async_wait0();
  __syncthreads();

  v2f ones; ones[0] = 1.f; ones[1] = 1.f;      // A = all-ones 16x4
  v8f c0 = {0.f, 0.f, 0.f, 0.f, 0.f, 0.f, 0.f, 0.f};
  v8f c1 = c0;
  float sq0 = 0.f, sq1 = 0.f;

  for (int t = 0; t < ntiles; ++t) {
    const float* cur = buf[t & 1];

    // prefetch tile t+1 into the other buffer while we consume tile t
    if (t + 1 < ntiles) {
      float* nxt = buf[(t + 1) & 1];
      const long long rowbase = off + (long long)(t + 1) * TILE_ROWS;
      for (int rr = 0; rr < rows_per_wave; ++rr) {
        int r = wave * rows_per_wave + rr;
        if ((t + 1) * TILE_ROWS + r < len)
          async_copy16(in + (rowbase + r) * (long long)DIMS + lane * 4,
                       nxt + r * DIMS + lane * 4);
      }
    }

    const int rows_left = len - t * TILE_ROWS;

    // B fragment base for this lane: rows (2*khalf, 2*khalf+1), column `col`.
    const float* bptr = cur + (2 * khalf) * DIMS + col;

    if (rows_left >= TILE_ROWS) {
      // ---- fast path: full tile, no guards, constant ds offsets ----------
#pragma unroll
      for (int k0 = 0; k0 < TILE_ROWS; k0 += 8) {
        float a0 = bptr[(k0 + 0) * DIMS];
        float a1 = bptr[(k0 + 1) * DIMS];
        float b0 = bptr[(k0 + 4) * DIMS];
        float b1 = bptr[(k0 + 5) * DIMS];
        c0 = wmma_ones_acc(ones, a0, a1, c0);
        c1 = wmma_ones_acc(ones, b0, b1, c1);
        sq0 = __builtin_fmaf(a0, a0, sq0);
        sq1 = __builtin_fmaf(a1, a1, sq1);
        sq0 = __builtin_fmaf(b0, b0, sq0);
        sq1 = __builtin_fmaf(b1, b1, sq1);
      }
    } else {
      // ---- guarded tail tile: zero-pad beyond rows_left (cndmask) --------
#pragma unroll
      for (int k0 = 0; k0 < TILE_ROWS; k0 += 4) {
        int r0 = k0 + 2 * khalf;
        float t0 = bptr[k0 * DIMS];
        float t1 = bptr[k0 * DIMS + DIMS];
        float b0 = (r0     < rows_left) ? t0 : 0.f;
        float b1 = (r0 + 1 < rows_left) ? t1 : 0.f;
        c0 = wmma_ones_acc(ones, b0, b1, c0);
        sq0 = __builtin_fmaf(b0, b0, sq0);
        sq1 = __builtin_fmaf(b1, b1, sq1);
      }
    }

    async_wait0();      // prefetch landed before anyone touches the other buffer
    __syncthreads();    // everyone done reading `cur` before it is re-staged
  }

  // combine accumulators (fixed order -> deterministic)
  const float colsum = c0[0] + c1[0];  // every D row holds the column sums
  const float sq = sq0 + sq1;

  if (lane < 16) red_col[col] = colsum;  // lanes 0-15 carry N=0..15
  red_sq[tid] = sq;
  __syncthreads();

  if (tid < DIMS) { float v = red_col[tid]; red_col[tid] = v * v; }
  __syncthreads();
  for (int s = DIMS / 2; s > 0; s >>= 1) {      // fixed-order tree: deterministic
    if (tid < s) red_col[tid] += red_col[tid + s];
    __syncthreads();
  }
  for (int s = 128; s > 0; s >>= 1) {
    if (tid < s) red_sq[tid] += red_sq[tid + s];
    __syncthreads();
  }
  if (tid == 0) segval[seg] = red_sq[0] - red_col[0] / (float)len;
}

// ---- kernel 3: 2048 -> 1 deterministic reduce -----------------------------
__global__ __launch_bounds__(256) void seg_finish_kernel(
    const float* __restrict__ segval, int S, float* __restrict__ out,
    float inv_nd) {
  __shared__ float sh[256];
  const int t = threadIdx.x;
  const int chunk = (S + 255) / 256;
  float a = 0.f;
  for (int j = 0; j < chunk; ++j) {
    int i = t * chunk + j;
    if (i < S) a += segval[i];
  }
  sh[t] = a;
  __syncthreads();
  for (int s = 128; s > 0; s >>= 1) {
    if (t < s) sh[t] += sh[t + s];
    __syncthreads();
  }
  if (t == 0) out[0] = sh[0] * inv_nd;
}

// ---------------------------------------------------------------------------
extern "C" void kernel_launch(void* const* d_in, const int* in_sizes, int n_in,
                              void* d_out, int out_size, void* d_ws, size_t ws_size,
                              hipStream_t stream) {
  (void)n_in; (void)out_size; (void)ws_size;
  const float* in = (const float*)d_in[0];            // [N, 128] f32
  const long long* lens = (const long long*)d_in[1];  // [S] int64
  const int S = in_sizes[1];
  const long long ND = (long long)in_sizes[0];        // N * D

  int* offs = (int*)d_ws;
  size_t off_bytes = ((size_t)S * sizeof(int) + 255u) & ~(size_t)255u;
  float* segval = (float*)((char*)d_ws + off_bytes);

  seg_scan_kernel<<<1, 256, 0, stream>>>(lens, S, offs);
  seg_mse_kernel<<<S, 256, 0, stream>>>(in, lens, offs, segval);
  seg_finish_kernel<<<1, 256, 0, stream>>>(segval, S, (float*)d_out,
                                           (float)(1.0 / (double)ND));
}